// DeepFM_65970697666829
// MI455X (gfx1250) — compile-verified
//
#include <hip/hip_runtime.h>
#include <hip/hip_bf16.h>
#include <stdint.h>

typedef __attribute__((ext_vector_type(2))) float v2f;
typedef __attribute__((ext_vector_type(8))) float v8f;

#define BN_EPS 1e-5f

__device__ inline v8f wmma_f32_16x16x4(v2f a, v2f b, v8f c) {
#if defined(__gfx1250__)
    // 8 args: (neg_a, A, neg_b, B, c_mod, C, reuse_a, reuse_b)
    return __builtin_amdgcn_wmma_f32_16x16x4_f32(
        false, a, false, b, (short)0, c, false, false);
#else
    return c;  // host pass placeholder
#endif
}

// ---------------------------------------------------------------------------
// CDNA5 async global->LDS staging (ASYNCcnt-tracked, no VGPR round trip).
// Builtin signature (from hipcc diagnostic): params are pointers to
// int __attribute__((vector_size(16))) in AS1 (global) / AS3 (LDS).
// Falls back to a synchronous copy if the builtins are unavailable.
// ---------------------------------------------------------------------------
#if defined(__gfx1250__) && \
    __has_builtin(__builtin_amdgcn_global_load_async_to_lds_b128) && \
    __has_builtin(__builtin_amdgcn_s_wait_asynccnt)
#define USE_ASYNC_G2L 1
#else
#define USE_ASYNC_G2L 0
#endif

typedef int v4i __attribute__((vector_size(16)));
typedef __attribute__((address_space(1))) v4i as1_v4i;
typedef __attribute__((address_space(3))) v4i as3_v4i;

__device__ __forceinline__ void async_copy_b128(const float* g, float* l) {
#if USE_ASYNC_G2L
    // generic LDS pointer low 32 bits == LDS byte offset (ISA 10.2 aperture)
    __builtin_amdgcn_global_load_async_to_lds_b128(
        (as1_v4i*)(uintptr_t)g,
        (as3_v4i*)(uint32_t)(uintptr_t)l,
        /*imm offset*/ 0, /*cpol*/ 0);
#else
    *(float4*)l = *(const float4*)g;
#endif
}
__device__ __forceinline__ void async_wait_3() {
#if USE_ASYNC_G2L
    __builtin_amdgcn_s_wait_asynccnt(3);
#endif
}
__device__ __forceinline__ void async_wait_0() {
#if USE_ASYNC_G2L
    __builtin_amdgcn_s_wait_asynccnt(0);
#endif
}

// ---------------------------------------------------------------------------
// Kernel 1: embeddings + FM first/second order.
// Block = 256 threads = 16 samples x 16 embedding lanes.
// Writes X (N x 576) = fm_first, and base[n] = sum(fm_first)+sum(fm_second)+bias.
// ---------------------------------------------------------------------------
__global__ __launch_bounds__(256) void embed_fm_kernel(
    const float* __restrict__ Xi_dense,  // N x 26
    const float* __restrict__ Xv,        // N x 36
    const float* __restrict__ Wd,        // 26 x 16
    const float* __restrict__ bd,        // 26 x 16
    const float* __restrict__ tables,    // 10 x 200000 x 16
    const int*   __restrict__ Xi_cat,    // N x 10
    const float* __restrict__ bias_vec,  // N
    float* __restrict__ Xout,            // N x 576
    float* __restrict__ base,            // N
    int N)
{
    const int t  = threadIdx.x;
    const int e  = t & 15;
    const int nl = t >> 4;
    const int n  = blockIdx.x * 16 + nl;
    if (n >= N) return;

    const float* xv = Xv + (size_t)n * 36;
    float*       xo = Xout + (size_t)n * 576;

    float s = 0.f, ss = 0.f, first = 0.f;

    #pragma unroll 2
    for (int f = 0; f < 26; ++f) {
        float v = Xi_dense[(size_t)n * 26 + f] * Wd[f * 16 + e] + bd[f * 16 + e];
        v *= xv[f];
        xo[f * 16 + e] = v;
        s += v; ss += v * v; first += v;
    }
    #pragma unroll 2
    for (int f = 0; f < 10; ++f) {
        int idx = Xi_cat[(size_t)n * 10 + f];
        float v = tables[((size_t)f * 200000 + (size_t)idx) * 16 + e];
        v *= xv[26 + f];
        xo[(26 + f) * 16 + e] = v;
        s += v; ss += v * v; first += v;
    }
    float tot = first + 0.5f * (s * s - ss);
    // reduce across the 16 embedding lanes (groups of 16 within wave32)
    tot += __shfl_xor(tot, 8, 16);
    tot += __shfl_xor(tot, 4, 16);
    tot += __shfl_xor(tot, 2, 16);
    tot += __shfl_xor(tot, 1, 16);
    if (e == 0) base[n] = tot + bias_vec[n];
}

// ---------------------------------------------------------------------------
// Kernel 2/5: WMMA f32 GEMM  Y = X @ W^T + bias
//   X: M x K (row-major), W: Nw x K (row-major), Y: M x Nw.
// Block tile 128x64, 8 waves (4x2), each wave 32x32 = 2x2 WMMA 16x16 tiles.
// K staged 16-wide, DOUBLE-BUFFERED via global_load_async_to_lds_b128:
// the async engine fetches stage k+1 while the wave runs the WMMA clause on
// stage k.  LDS rows padded to 20 floats -> conflict-free ds_load_b64 and
// 16B-aligned b128 async writes.
// Requires M % 128 == 0, K % 16 == 0.
// ---------------------------------------------------------------------------
#define GTM 128
#define GTN 64
#define GTK 16
#define LDP 20                 // padded LDS row stride (floats)
#define ASZ (GTM * LDP)        // 2560 floats per A buffer
#define BSZ (GTN * LDP)        // 1280 floats per B buffer

__global__ __launch_bounds__(256) void wmma_gemm_bias_kernel(
    const float* __restrict__ X, const float* __restrict__ W,
    const float* __restrict__ bias, float* __restrict__ Y,
    int M, int Nw, int K)
{
    __shared__ float As[2 * ASZ];  // 20480 B
    __shared__ float Bs[2 * BSZ];  // 10240 B

    const int t    = threadIdx.x;
    const int wave = t >> 5;
    const int lane = t & 31;
    const int wm   = wave & 3;   // wave m position (0..3)
    const int wn   = wave >> 2;  // wave n position (0..1)
    const int l16  = lane & 15;
    const int half = lane >> 4;  // selects K-pair (0,1) vs (2,3) per ISA layout

    const int m0 = blockIdx.x * GTM;
    const int n0 = blockIdx.y * GTN;

    // staging coordinates: A = 128x16 floats (2 x b128/thread),
    //                      B =  64x16 floats (1 x b128/thread)
    const int lin0  = t * 4;
    const int arow0 = lin0 >> 4, acol0 = lin0 & 15;
    const int lin1  = lin0 + 1024;
    const int arow1 = lin1 >> 4, acol1 = lin1 & 15;
    const int brow  = arow0;     // 0..63
    const int bcol  = acol0;
    int bn = n0 + brow;
    if (bn >= Nw) bn = Nw - 1;   // clamp: data lands in never-stored accums

    const float* gA0 = X + (size_t)(m0 + arow0) * K + acol0;
    const float* gA1 = X + (size_t)(m0 + arow1) * K + acol1;
    const float* gB  = W + (size_t)bn * K + bcol;
    float* lA0 = As + arow0 * LDP + acol0;
    float* lA1 = As + arow1 * LDP + acol1;
    float* lB  = Bs + brow  * LDP + bcol;

    v8f acc[2][2] = {};
    const int nst = K / GTK;

    // prologue: stage 0 into buffer 0
    async_copy_b128(gA0, lA0);
    async_copy_b128(gA1, lA1);
    async_copy_b128(gB,  lB);

    for (int ks = 0; ks < nst; ++ks) {
        if (ks + 1 < nst) {
            // issue stage ks+1 into the other buffer, then wait for stage ks
            const int k1 = (ks + 1) * GTK;
            const int p1 = ((ks + 1) & 1);
            async_copy_b128(gA0 + k1, lA0 + p1 * ASZ);
            async_copy_b128(gA1 + k1, lA1 + p1 * ASZ);
            async_copy_b128(gB  + k1, lB  + p1 * BSZ);
            async_wait_3();          // <=3 may remain in flight (next stage)
        } else {
            async_wait_0();          // last stage: drain everything
        }
        __syncthreads();             // stage ks visible to all waves

        const float* a = As + (ks & 1) * ASZ;
        const float* b = Bs + (ks & 1) * BSZ;
        #pragma unroll
        for (int kk = 0; kk < GTK; kk += 4) {
            const int c = kk + 2 * half;
            v2f a0 = *(const v2f*)(&a[(wm * 32      + l16) * LDP + c]);
            v2f a1 = *(const v2f*)(&a[(wm * 32 + 16 + l16) * LDP + c]);
            v2f b0 = *(const v2f*)(&b[(wn * 32      + l16) * LDP + c]);
            v2f b1 = *(const v2f*)(&b[(wn * 32 + 16 + l16) * LDP + c]);
            acc[0][0] = wmma_f32_16x16x4(a0, b0, acc[0][0]);
            acc[0][1] = wmma_f32_16x16x4(a0, b1, acc[0][1]);
            acc[1][0] = wmma_f32_16x16x4(a1, b0, acc[1][0]);
            acc[1][1] = wmma_f32_16x16x4(a1, b1, acc[1][1]);
        }
        __syncthreads();             // all waves done reading before reload
    }

    // epilogue: D layout -> VGPR r : M = r + 8*half ; N = lane&15
    #pragma unroll
    for (int i = 0; i < 2; ++i) {
        const int mbase = m0 + wm * 32 + i * 16 + half * 8;
        #pragma unroll
        for (int j = 0; j < 2; ++j) {
            const int n = n0 + wn * 32 + j * 16 + l16;
            if (n < Nw) {
                const float bn2 = bias[n];
                #pragma unroll
                for (int r = 0; r < 8; ++r) {
                    Y[(size_t)(mbase + r) * Nw + n] = acc[i][j][r] + bn2;
                }
            }
        }
    }
}

// ---------------------------------------------------------------------------
// Kernel 3/6: per-column BN-train stats -> folded affine a[h], c[h]:
//   a = gamma * rsqrt(var + eps),  c = beta - mu * a
// Block = 16 cols x 16 row-threads; deterministic fixed-order reduction.
// ---------------------------------------------------------------------------
__global__ __launch_bounds__(256) void bn_stats_kernel(
    const float* __restrict__ Y, const float* __restrict__ g,
    const float* __restrict__ be, float* __restrict__ a,
    float* __restrict__ c, int N, int H)
{
    __shared__ float s1[256];
    __shared__ float s2[256];
    const int t   = threadIdx.x;
    const int cl  = t & 15;
    const int j   = t >> 4;
    const int col = blockIdx.x * 16 + cl;

    float sum = 0.f, ssq = 0.f;
    if (col < H) {
        for (int r = j; r < N; r += 16) {
            float v = Y[(size_t)r * H + col];
            sum += v; ssq += v * v;
        }
    }
    s1[t] = sum; s2[t] = ssq;
    __syncthreads();
    if (t < 16) {
        const int colw = blockIdx.x * 16 + t;
        float S = 0.f, Q = 0.f;
        #pragma unroll
        for (int k = 0; k < 16; ++k) { S += s1[t + k * 16]; Q += s2[t + k * 16]; }
        if (colw < H) {
            float inv_n = 1.0f / (float)N;
            float mu  = S * inv_n;
            float var = Q * inv_n - mu * mu;
            float av  = g[colw] * rsqrtf(var + BN_EPS);
            a[colw] = av;
            c[colw] = be[colw] - mu * av;
        }
    }
}

// ---------------------------------------------------------------------------
// Kernel 4: fold BN1 into layer-2 weights:
//   W2p[j,h] = W2[j,h]*a1[h];  b2p[j] = b2[j] + sum_h c1[h]*W2[j,h]
// One block per output row j.
// ---------------------------------------------------------------------------
__global__ __launch_bounds__(256) void fold_w2_kernel(
    const float* __restrict__ W2, const float* __restrict__ b2,
    const float* __restrict__ a1, const float* __restrict__ c1,
    float* __restrict__ W2p, float* __restrict__ b2p, int H)
{
    __shared__ float red[256];
    const int j = blockIdx.x;
    const int t = threadIdx.x;
    float part = 0.f;
    for (int h = t; h < H; h += 256) {
        float w = W2[(size_t)j * H + h];
        W2p[(size_t)j * H + h] = w * a1[h];
        part += w * c1[h];
    }
    red[t] = part;
    __syncthreads();
    for (int off = 128; off > 0; off >>= 1) {
        if (t < off) red[t] += red[t + off];
        __syncthreads();
    }
    if (t == 0) b2p[j] = b2[j] + red[0];
}

// ---------------------------------------------------------------------------
// Kernel 7: out[n] = base[n] + sum_h (Y2[n,h]*a2[h] + c2[h])
// Block = 16 samples x 16 col-threads.
// ---------------------------------------------------------------------------
__global__ __launch_bounds__(256) void final_kernel(
    const float* __restrict__ Y2, const float* __restrict__ a2,
    const float* __restrict__ c2, const float* __restrict__ base,
    float* __restrict__ out, int N, int H)
{
    __shared__ float red[256];
    const int t  = threadIdx.x;
    const int e  = t & 15;
    const int nl = t >> 4;
    const int n  = blockIdx.x * 16 + nl;

    float acc = 0.f;
    if (n < N) {
        for (int h = e; h < H; h += 16) {
            acc += Y2[(size_t)n * H + h] * a2[h] + c2[h];
        }
    }
    red[t] = acc;
    __syncthreads();
    if (e == 0 && n < N) {
        float s = 0.f;
        #pragma unroll
        for (int k = 0; k < 16; ++k) s += red[nl * 16 + k];
        out[n] = base[n] + s;
    }
}

// ---------------------------------------------------------------------------
extern "C" void kernel_launch(void* const* d_in, const int* in_sizes, int n_in,
                              void* d_out, int out_size, void* d_ws, size_t ws_size,
                              hipStream_t stream) {
    const float* Xi_dense = (const float*)d_in[0];
    const float* Xv       = (const float*)d_in[1];
    const float* Wd       = (const float*)d_in[2];
    const float* bd       = (const float*)d_in[3];
    const float* tables   = (const float*)d_in[4];
    const float* W1       = (const float*)d_in[5];
    const float* b1       = (const float*)d_in[6];
    const float* g1       = (const float*)d_in[7];
    const float* be1      = (const float*)d_in[8];
    const float* W2       = (const float*)d_in[9];
    const float* b2       = (const float*)d_in[10];
    const float* g2       = (const float*)d_in[11];
    const float* be2      = (const float*)d_in[12];
    const float* bias_vec = (const float*)d_in[13];
    const int*   Xi_cat   = (const int*)d_in[14];

    const int N = 16384, H = 400, DIN = 576;

    // workspace carving (256B aligned)
    char* ws = (char*)d_ws;
    size_t off = 0;
    auto carve = [&](size_t bytes) -> void* {
        void* p = ws + off;
        off += (bytes + 255) & ~(size_t)255;
        return p;
    };
    float* X    = (float*)carve((size_t)N * DIN * sizeof(float));   // fm_first
    float* Y1   = (float*)carve((size_t)N * H * sizeof(float));
    float* Y2   = (float*)carve((size_t)N * H * sizeof(float));
    float* base = (float*)carve((size_t)N * sizeof(float));
    float* a1   = (float*)carve((size_t)H * sizeof(float));
    float* c1   = (float*)carve((size_t)H * sizeof(float));
    float* a2   = (float*)carve((size_t)H * sizeof(float));
    float* c2   = (float*)carve((size_t)H * sizeof(float));
    float* W2p  = (float*)carve((size_t)H * H * sizeof(float));
    float* b2p  = (float*)carve((size_t)H * sizeof(float));
    float* out  = (float*)d_out;

    // 1. embeddings + FM  -> X, base
    embed_fm_kernel<<<N / 16, 256, 0, stream>>>(
        Xi_dense, Xv, Wd, bd, tables, Xi_cat, bias_vec, X, base, N);

    // 2. Y1 = X @ W1^T + b1        (M=16384, Nw=400, K=576)
    {
        dim3 grid(N / GTM, (H + GTN - 1) / GTN);
        wmma_gemm_bias_kernel<<<grid, 256, 0, stream>>>(X, W1, b1, Y1, N, H, DIN);
    }

    // 3. BN1 stats -> a1, c1
    bn_stats_kernel<<<(H + 15) / 16, 256, 0, stream>>>(Y1, g1, be1, a1, c1, N, H);

    // 4. fold BN1 into W2 -> W2p, b2p
    fold_w2_kernel<<<H, 256, 0, stream>>>(W2, b2, a1, c1, W2p, b2p, H);

    // 5. Y2 = Y1 @ W2p^T + b2p     (M=16384, Nw=400, K=400)
    {
        dim3 grid(N / GTM, (H + GTN - 1) / GTN);
        wmma_gemm_bias_kernel<<<grid, 256, 0, stream>>>(Y1, W2p, b2p, Y2, N, H, H);
    }

    // 6. BN2 stats -> a2, c2
    bn_stats_kernel<<<(H + 15) / 16, 256, 0, stream>>>(Y2, g2, be2, a2, c2, N, H);

    // 7. out[n] = base[n] + sum_h (Y2[n,h]*a2[h] + c2[h])
    final_kernel<<<N / 16, 256, 0, stream>>>(Y2, a2, c2, base, out, N, H);
}